// Attention_63909113364946
// MI455X (gfx1250) — compile-verified
//
#include <hip/hip_runtime.h>
#include <hip/hip_bf16.h>
#include <cstdint>
#include <cstddef>

#define B_   64
#define T_   2048
#define RNN_ 1024
#define EMB_ 512
#define ATT_ 128
#define NF_  32
#define KSZ_ 31

typedef __attribute__((ext_vector_type(2))) float v2f;
typedef __attribute__((ext_vector_type(8))) float v8f;

// gfx1250 has a hardware tanh (v_tanh_f32) — avoids the branchy __ocml expansion
__device__ __forceinline__ float fast_tanh(float x) {
#if __has_builtin(__builtin_amdgcn_tanhf)
    return __builtin_amdgcn_tanhf(x);
#else
    return tanhf(x);
#endif
}

// ---------------------------------------------------------------------------
// Kernel 1: processed_query  pq[b][a] = sum_k hidden[b][k] * Wq[a][k]
// ---------------------------------------------------------------------------
__global__ __launch_bounds__(128)
void pq_kernel(const float* __restrict__ h,      // [B][RNN]
               const float* __restrict__ Wq,     // [ATT][RNN]
               float* __restrict__ pq)           // [B][ATT]
{
    __shared__ float sh[RNN_];
    const int b = blockIdx.x;
    for (int i = threadIdx.x; i < RNN_; i += 128) sh[i] = h[b * RNN_ + i];
    __syncthreads();
    const int a = threadIdx.x;
    const float* w = Wq + (size_t)a * RNN_;
    float acc = 0.f;
    #pragma unroll 8
    for (int k = 0; k < RNN_; ++k) acc = fmaf(sh[k], w[k], acc);
    pq[b * ATT_ + a] = acc;
}

// ---------------------------------------------------------------------------
// Kernel 2: 'same' conv1d  loc[b][t][f] = sum_{c,k} cat[b][c][t-15+k]*Wc[f][c][k]
// ---------------------------------------------------------------------------
__global__ __launch_bounds__(256)
void conv_kernel(const float* __restrict__ cat,  // [B][2][T]
                 const float* __restrict__ Wc,   // [NF][2][KSZ]
                 float* __restrict__ loc)        // [B][T][NF]
{
    const int i = blockIdx.x * 256 + threadIdx.x;   // i = b*65536 + t*32 + f
    const int f = i & 31;
    const int t = (i >> 5) & (T_ - 1);
    const int b = i >> 16;
    const float* c0 = cat + (size_t)(b * 2 + 0) * T_;
    const float* c1 = cat + (size_t)(b * 2 + 1) * T_;
    const float* w  = Wc + (size_t)f * (2 * KSZ_);
    float acc = 0.f;
    #pragma unroll
    for (int k = 0; k < KSZ_; ++k) {
        const int tt = t - 15 + k;
        if (tt >= 0 && tt < T_) {
            acc = fmaf(c0[tt], w[k], acc);
            acc = fmaf(c1[tt], w[KSZ_ + k], acc);
        }
    }
    loc[i] = acc;
}

// ---------------------------------------------------------------------------
// Kernel 3 (WMMA): energies[b][t] = sum_a Wv[a]*tanh(pq[b][a]+ploc[b][t][a]+pin[b][t][a]) + bv
// ploc tile (16t x 16a) computed with V_WMMA_F32_16X16X4_F32, K=f reduced in 8 steps.
// One wave32 per 16-row t-tile; 8 waves (256 thr) per block.
// ---------------------------------------------------------------------------
__global__ __launch_bounds__(256)
void energies_kernel(const float* __restrict__ loc,  // [B][T][NF]
                     const float* __restrict__ pin,  // [B][T][ATT]
                     const float* __restrict__ pq,   // [B][ATT]
                     const float* __restrict__ Wl,   // [ATT][NF]
                     const float* __restrict__ Wv,   // [ATT]
                     const float* __restrict__ bv,   // [1]
                     float* __restrict__ energies)   // [B][T]
{
    __shared__ float sWl[ATT_ * NF_];                // 16 KB
    for (int i = threadIdx.x; i < ATT_ * NF_; i += 256) sWl[i] = Wl[i];
    __syncthreads();

    const int lane = threadIdx.x & 31;
    const int wave = threadIdx.x >> 5;
    const int tile = blockIdx.x * 8 + wave;          // global 16-row t-tile id
    const int b    = tile >> 7;                      // 128 tiles per batch row
    const int t0   = (tile & 127) << 4;

    const int m  = lane & 15;                        // row-in-tile / N index
    const int hi = lane >> 4;                        // K-half select

    // A fragments: A[M=m][K = 4*kk + 2*hi + {0,1}] = loc[b][t0+m][f]
    const float* locRow = loc + ((size_t)(b * T_ + t0 + m) * NF_);
    float aBuf[16];
    #pragma unroll
    for (int kk = 0; kk < 8; ++kk) {
        aBuf[2 * kk + 0] = locRow[4 * kk + 2 * hi + 0];
        aBuf[2 * kk + 1] = locRow[4 * kk + 2 * hi + 1];
    }

    float epart[8];
    #pragma unroll
    for (int v = 0; v < 8; ++v) epart[v] = 0.f;

    for (int na = 0; na < 8; ++na) {                 // N (attention-dim) tiles
        const int a0 = na << 4;
        v8f acc = {};                                // C = 0
        #pragma unroll
        for (int kk = 0; kk < 8; ++kk) {             // K steps of 4 over f=32
            v2f av = { aBuf[2 * kk + 0], aBuf[2 * kk + 1] };
            // B[K = 4*kk + 2*hi + v][N=m] = Wl[a0+m][f]
            const float* wrow = &sWl[(a0 + m) * NF_ + 4 * kk + 2 * hi];
            v2f bf = { wrow[0], wrow[1] };
            acc = __builtin_amdgcn_wmma_f32_16x16x4_f32(
                false, av, false, bf, (short)0, acc, false, false);
        }
        // D element v: row t = t0 + v + 8*hi, col a = a0 + m
        const float pqv = pq[b * ATT_ + a0 + m];
        const float wvv = Wv[a0 + m];
        #pragma unroll
        for (int v = 0; v < 8; ++v) {
            const int t = t0 + v + 8 * hi;
            const float x = acc[v] + pqv + pin[((size_t)(b * T_ + t) * ATT_) + a0 + m];
            epart[v] += wvv * fast_tanh(x);
        }
    }

    // reduce over the 16 N-lanes within each half-wave
    #pragma unroll
    for (int v = 0; v < 8; ++v) {
        float e = epart[v];
        #pragma unroll
        for (int s = 1; s < 16; s <<= 1) e += __shfl_xor(e, s, 32);
        epart[v] = e;
    }
    if (m == 0) {
        const float bias = bv[0];
        #pragma unroll
        for (int v = 0; v < 8; ++v)
            energies[b * T_ + t0 + v + 8 * hi] = epart[v] + bias;
    }
}

// ---------------------------------------------------------------------------
// Kernel 4: masked softmax over T; writes both alignment output copies
// ---------------------------------------------------------------------------
__global__ __launch_bounds__(256)
void softmax_kernel(const float* __restrict__ e,          // [B][T]
                    const uint8_t* __restrict__ mask,     // [B][T] (bool)
                    float* __restrict__ out1,             // [B][T]
                    float* __restrict__ out2)             // [B][T]
{
    __shared__ float red[256];
    const int b = blockIdx.x;
    const int tid = threadIdx.x;

    float vals[8];
    float mx = -INFINITY;
    #pragma unroll
    for (int j = 0; j < 8; ++j) {
        const int t = tid + j * 256;
        float v = e[b * T_ + t];
        v = mask[b * T_ + t] ? v : -INFINITY;
        vals[j] = v;
        mx = fmaxf(mx, v);
    }
    red[tid] = mx; __syncthreads();
    for (int s = 128; s > 0; s >>= 1) {
        if (tid < s) red[tid] = fmaxf(red[tid], red[tid + s]);
        __syncthreads();
    }
    mx = red[0]; __syncthreads();

    float sum = 0.f;
    #pragma unroll
    for (int j = 0; j < 8; ++j) { vals[j] = __expf(vals[j] - mx); sum += vals[j]; }
    red[tid] = sum; __syncthreads();
    for (int s = 128; s > 0; s >>= 1) {
        if (tid < s) red[tid] += red[tid + s];
        __syncthreads();
    }
    const float inv = 1.f / red[0];

    #pragma unroll
    for (int j = 0; j < 8; ++j) {
        const int t = tid + j * 256;
        const float a = vals[j] * inv;
        out1[b * T_ + t] = a;
        out2[b * T_ + t] = a;
    }
}

// ---------------------------------------------------------------------------
// Kernel 5: context[b][d] = sum_t align[b][t] * inputs[b][t][d]
// Bandwidth-critical: 256 MB of `inputs`, fully coalesced, touched once.
// ---------------------------------------------------------------------------
__global__ __launch_bounds__(256)
void context_kernel(const float* __restrict__ align,  // [B][T]
                    const float* __restrict__ x,      // [B][T][EMB]
                    float* __restrict__ ctx)          // [B][EMB]
{
    __shared__ float sa[256];
    const int b = blockIdx.x;
    const int d = blockIdx.y * 256 + threadIdx.x;
    float acc = 0.f;
    for (int tc = 0; tc < T_; tc += 256) {
        __syncthreads();
        sa[threadIdx.x] = align[b * T_ + tc + threadIdx.x];
        __syncthreads();
        #pragma unroll 4
        for (int j = 0; j < 256; ++j)
            acc = fmaf(sa[j], x[((size_t)(b * T_ + tc + j) * EMB_) + d], acc);
    }
    ctx[b * EMB_ + d] = acc;
}

// ---------------------------------------------------------------------------
extern "C" void kernel_launch(void* const* d_in, const int* in_sizes, int n_in,
                              void* d_out, int out_size, void* d_ws, size_t ws_size,
                              hipStream_t stream) {
    const float*   hidden = (const float*)d_in[0];    // [B, RNN]
    const float*   inputs = (const float*)d_in[1];    // [B, T, EMB]
    const float*   pin    = (const float*)d_in[2];    // [B, T, ATT]
    const float*   cat    = (const float*)d_in[3];    // [B, 2, T]
    const uint8_t* mask   = (const uint8_t*)d_in[4];  // [B, T] bool
    const float*   Wq     = (const float*)d_in[5];    // [ATT, RNN]
    const float*   Wconv  = (const float*)d_in[6];    // [NF, 2, KSZ]
    const float*   Wl     = (const float*)d_in[7];    // [ATT, NF]
    const float*   Wv     = (const float*)d_in[8];    // [1, ATT]
    const float*   bv     = (const float*)d_in[9];    // [1]

    float* ctx    = (float*)d_out;                    // [B, EMB]
    float* align1 = ctx + B_ * EMB_;                  // [B, T]
    float* align2 = align1 + B_ * T_;                 // [B, T]

    // workspace layout
    float* loc_ws = (float*)d_ws;                     // [B, T, NF]   16 MB
    float* pq_ws  = loc_ws + (size_t)B_ * T_ * NF_;   // [B, ATT]     32 KB
    float* e_ws   = pq_ws + (size_t)B_ * ATT_;        // [B, T]       512 KB

    pq_kernel<<<B_, 128, 0, stream>>>(hidden, Wq, pq_ws);
    conv_kernel<<<(B_ * T_ * NF_) / 256, 256, 0, stream>>>(cat, Wconv, loc_ws);
    energies_kernel<<<(B_ * T_ / 16) / 8, 256, 0, stream>>>(loc_ws, pin, pq_ws,
                                                            Wl, Wv, bv, e_ws);
    softmax_kernel<<<B_, 256, 0, stream>>>(e_ws, mask, align1, align2);
    context_kernel<<<dim3(B_, EMB_ / 256), 256, 0, stream>>>(align1, inputs, ctx);
}